// GIN_7602092113945
// MI455X (gfx1250) — compile-verified
//
#include <hip/hip_runtime.h>
#include <hip/hip_bf16.h>
#include <math.h>

// Fixed problem dimensions from setup_inputs()
#define NN0 135168
#define NN1 12288
#define NN2 2048
#define EE1 122880
#define EE2 10240
#define F_IN 602
#define F_PAD 604   // K padded to multiple of 4 for 16x16x4 f32 WMMA
#define H_DIM 256
#define C_DIM 41
#define C_PAD 64    // N padded to multiple of 32 (wave computes two 16-wide N tiles)

typedef __attribute__((ext_vector_type(2))) float v2f;
typedef __attribute__((ext_vector_type(8))) float v8f;

// Generic zero-padding copy: dst[r,c] = (r<srcR && c<srcC) ? src[r,c] : 0
__global__ void gin_pad2d_kernel(const float* __restrict__ src, float* __restrict__ dst,
                                 int srcR, int srcC, int dstR, int dstC) {
  long total = (long)dstR * dstC;
  for (long i = blockIdx.x * (long)blockDim.x + threadIdx.x; i < total;
       i += (long)gridDim.x * blockDim.x) {
    int r = (int)(i / dstC);
    int c = (int)(i % dstC);
    dst[i] = (r < srcR && c < srcC) ? src[(long)r * srcC + c] : 0.0f;
  }
}

// Segment-sum scatter: one block per edge; agg[dst[e], :] += xin[src[e], :]
// unsafeAtomicAdd -> native global_atomic_add_f32 (no CAS retry loop)
__global__ void gin_scatter_kernel(const float* __restrict__ xin, const int* __restrict__ src,
                                   const int* __restrict__ dst, float* __restrict__ agg,
                                   int F, int ldagg) {
  int e = blockIdx.x;
  int s = src[e];
  int d = dst[e];
  const float* xr = xin + (long)s * F;
  float* ar = agg + (long)d * ldagg;
  for (int f = threadIdx.x; f < F; f += blockDim.x)
    unsafeAtomicAdd(ar + f, xr[f]);
}

// WMMA fp32 GEMM: C[M,N] = epilogue(A[M,K] @ B[K,N])
// Each wave computes a 16x32 output strip = two 16x16 WMMA tiles sharing the A fragment.
// K consumed 4 at a time via v_wmma_f32_16x16x4_f32. ntn (N/16) must be even.
// mode: 0 = bias+BN+ReLU, 1 = bias+ReLU, 2 = bias only
__global__ void gin_wmma_gemm_kernel(const float* __restrict__ A, int lda,
                                     const float* __restrict__ B, int ldb,
                                     const float* __restrict__ bias, int Nreal,
                                     const float* __restrict__ g, const float* __restrict__ be,
                                     const float* __restrict__ mu, const float* __restrict__ var,
                                     float* __restrict__ C, int ldc,
                                     int K, int ntn, int mode) {
  int wave = threadIdx.x >> 5;          // wave32
  int lane = threadIdx.x & 31;
  int ntnp = ntn >> 1;                  // N-tile pairs
  int pair = blockIdx.x * (blockDim.x >> 5) + wave;  // grids launched exactly -> no guard
  int tm = pair / ntnp;
  int tnp = pair % ntnp;
  int half = lane >> 4;                 // 0: K={k,k+1}; 1: K={k+2,k+3}
  int l16 = lane & 15;
  int m = tm * 16 + l16;                // A row for this lane
  int n0 = tnp * 32 + l16;              // first N column
  int n1 = n0 + 16;                     // second N column

  v8f acc0 = {0.f, 0.f, 0.f, 0.f, 0.f, 0.f, 0.f, 0.f};
  v8f acc1 = {0.f, 0.f, 0.f, 0.f, 0.f, 0.f, 0.f, 0.f};

  // Strength-reduced pointers: no multiplies inside the K loop
  const float* Ap = A + (long)m * lda + half * 2;
  const float* B0 = B + (long)(half * 2) * ldb + n0;
  const float* B1 = B0 + 16;
  long bstep = 4L * ldb;

  for (int k = 0; k < K; k += 4) {
    v2f a = *(const v2f*)Ap;            // A[m][k+2*half .. +1], 8B-aligned contiguous
    v2f b0, b1;
    b0.x = B0[0];                       // B[k+2*half  ][n0]
    b0.y = B0[ldb];                     // B[k+2*half+1][n0]
    b1.x = B1[0];
    b1.y = B1[ldb];
    acc0 = __builtin_amdgcn_wmma_f32_16x16x4_f32(false, a, false, b0, (short)0, acc0, false, false);
    acc1 = __builtin_amdgcn_wmma_f32_16x16x4_f32(false, a, false, b1, (short)0, acc1, false, false);
    Ap += 4;
    B0 += bstep;
    B1 += bstep;
  }

  // Epilogue: per-column bias / BN affine, then ReLU; C/D layout:
  // VGPR r -> row = 16*tm + r + 8*(lane>=16), col = n0 / n1
  float bv0 = (n0 < Nreal) ? bias[n0] : 0.0f;
  float bv1 = (n1 < Nreal) ? bias[n1] : 0.0f;
  float sc0 = 1.0f, sh0 = 0.0f, sc1 = 1.0f, sh1 = 0.0f;
  if (mode == 0) {
    sc0 = g[n0] * rsqrtf(var[n0] + 1e-5f);
    sh0 = be[n0] - mu[n0] * sc0;
    sc1 = g[n1] * rsqrtf(var[n1] + 1e-5f);
    sh1 = be[n1] - mu[n1] * sc1;
  }
  float* Crow = C + (long)(tm * 16 + half * 8) * ldc;
#pragma unroll
  for (int r = 0; r < 8; ++r) {
    float v0 = acc0[r] + bv0;
    float v1 = acc1[r] + bv1;
    if (mode == 0) { v0 = v0 * sc0 + sh0; v1 = v1 * sc1 + sh1; }
    if (mode != 2) { v0 = fmaxf(v0, 0.0f); v1 = fmaxf(v1, 0.0f); }
    Crow[(long)r * ldc + n0] = v0;
    Crow[(long)r * ldc + n1] = v1;
  }
}

// One wave per row; 41 classes over 32 lanes (lane handles c=lane and c=lane+32)
__global__ void gin_logsoftmax_kernel(const float* __restrict__ logits, float* __restrict__ out) {
  int wave = threadIdx.x >> 5;
  int lane = threadIdx.x & 31;
  int row = blockIdx.x * (blockDim.x >> 5) + wave;
  const float* lr = logits + (long)row * C_PAD;
  float v0 = (lane < C_DIM) ? lr[lane] : -3.0e38f;
  float v1 = (lane + 32 < C_DIM) ? lr[lane + 32] : -3.0e38f;
  float mx = fmaxf(v0, v1);
  for (int off = 16; off > 0; off >>= 1) mx = fmaxf(mx, __shfl_xor(mx, off, 32));
  float s = 0.0f;
  if (lane < C_DIM) s += __expf(v0 - mx);
  if (lane + 32 < C_DIM) s += __expf(v1 - mx);
  for (int off = 16; off > 0; off >>= 1) s += __shfl_xor(s, off, 32);
  float lse = __logf(s) + mx;
  if (lane < C_DIM) out[(long)row * C_DIM + lane] = v0 - lse;
  if (lane + 32 < C_DIM) out[(long)row * C_DIM + lane + 32] = v1 - lse;
}

extern "C" void kernel_launch(void* const* d_in, const int* in_sizes, int n_in,
                              void* d_out, int out_size, void* d_ws, size_t ws_size,
                              hipStream_t stream) {
  const float* x   = (const float*)d_in[0];
  const int* src1  = (const int*)d_in[1];
  const int* dst1  = (const int*)d_in[2];
  const int* src2  = (const int*)d_in[3];
  const int* dst2  = (const int*)d_in[4];
  // d_in[5]=n1, d_in[6]=n2 are fixed scalars; dimensions hardcoded above
  const float* W1a = (const float*)d_in[7];
  const float* b1a = (const float*)d_in[8];
  const float* g1  = (const float*)d_in[9];
  const float* be1 = (const float*)d_in[10];
  const float* m1  = (const float*)d_in[11];
  const float* v1  = (const float*)d_in[12];
  const float* W1b = (const float*)d_in[13];
  const float* b1b = (const float*)d_in[14];
  const float* W2a = (const float*)d_in[15];
  const float* b2a = (const float*)d_in[16];
  const float* g2  = (const float*)d_in[17];
  const float* be2 = (const float*)d_in[18];
  const float* m2  = (const float*)d_in[19];
  const float* v2  = (const float*)d_in[20];
  const float* W2b = (const float*)d_in[21];
  const float* b2b = (const float*)d_in[22];
  float* out = (float*)d_out;

  // Workspace layout (floats), ~60 MB total
  float* ws     = (float*)d_ws;
  float* aggpad = ws;                                  // NN1 x F_PAD
  float* W1apad = aggpad + (long)NN1 * F_PAD;          // F_PAD x H_DIM
  float* h1mid  = W1apad + (long)F_PAD * H_DIM;        // NN1 x H_DIM
  float* h1     = h1mid + (long)NN1 * H_DIM;           // NN1 x H_DIM
  float* h2in   = h1 + (long)NN1 * H_DIM;              // NN2 x H_DIM
  float* h2mid  = h2in + (long)NN2 * H_DIM;            // NN2 x H_DIM
  float* W2bpad = h2mid + (long)NN2 * H_DIM;           // H_DIM x C_PAD
  float* logits = W2bpad + (long)H_DIM * C_PAD;        // NN2 x C_PAD

  // Pad weights (zero-fill K pad rows of W1a, N pad cols of W2b)
  gin_pad2d_kernel<<<(F_PAD * H_DIM + 255) / 256, 256, 0, stream>>>(
      W1a, W1apad, F_IN, H_DIM, F_PAD, H_DIM);
  gin_pad2d_kernel<<<(H_DIM * C_PAD + 255) / 256, 256, 0, stream>>>(
      W2b, W2bpad, H_DIM, C_DIM, H_DIM, C_PAD);

  // Layer 1: agg = x[:NN1] (GIN eps = 0), zero pad cols; then scatter edges
  gin_pad2d_kernel<<<((long)NN1 * F_PAD + 255) / 256, 256, 0, stream>>>(
      x, aggpad, NN1, F_IN, NN1, F_PAD);
  gin_scatter_kernel<<<EE1, 256, 0, stream>>>(x, src1, dst1, aggpad, F_IN, F_PAD);

  // GEMM1: (NN1 x 604) @ (604 x 256) + b1a -> BN1 -> ReLU
  gin_wmma_gemm_kernel<<<(NN1 / 16) * (H_DIM / 32) / 8, 256, 0, stream>>>(
      aggpad, F_PAD, W1apad, H_DIM, b1a, H_DIM, g1, be1, m1, v1,
      h1mid, H_DIM, F_PAD, H_DIM / 16, 0);

  // GEMM2: (NN1 x 256) @ (256 x 256) + b1b -> ReLU
  gin_wmma_gemm_kernel<<<(NN1 / 16) * (H_DIM / 32) / 8, 256, 0, stream>>>(
      h1mid, H_DIM, W1b, H_DIM, b1b, H_DIM, nullptr, nullptr, nullptr, nullptr,
      h1, H_DIM, H_DIM, H_DIM / 16, 1);

  // Layer 2: h2in = h1[:NN2]; scatter edges
  gin_pad2d_kernel<<<((long)NN2 * H_DIM + 255) / 256, 256, 0, stream>>>(
      h1, h2in, NN2, H_DIM, NN2, H_DIM);
  gin_scatter_kernel<<<EE2, 256, 0, stream>>>(h1, src2, dst2, h2in, H_DIM, H_DIM);

  // GEMM3: (NN2 x 256) @ (256 x 256) + b2a -> BN2 -> ReLU
  gin_wmma_gemm_kernel<<<(NN2 / 16) * (H_DIM / 32) / 8, 256, 0, stream>>>(
      h2in, H_DIM, W2a, H_DIM, b2a, H_DIM, g2, be2, m2, v2,
      h2mid, H_DIM, H_DIM, H_DIM / 16, 0);

  // GEMM4: (NN2 x 256) @ (256 x 64pad) + b2b (no relu) -> logits (padded)
  gin_wmma_gemm_kernel<<<(NN2 / 16) * (C_PAD / 32) / 8, 256, 0, stream>>>(
      h2mid, H_DIM, W2bpad, C_PAD, b2b, C_DIM, nullptr, nullptr, nullptr, nullptr,
      logits, C_PAD, H_DIM, C_PAD / 16, 2);

  // log_softmax over 41 classes, one wave per row
  gin_logsoftmax_kernel<<<NN2 / 8, 256, 0, stream>>>(logits, out);
}